// Regressor2Stage_76776835383713
// MI455X (gfx1250) — compile-verified
//
#include <hip/hip_runtime.h>
#include <hip/hip_bf16.h>

typedef __attribute__((ext_vector_type(2))) float v2f;
typedef __attribute__((ext_vector_type(8))) float v8f;

#define NH 448
#define NW 608
#define NC 64
#define NS1 32
#define NS2 16
#define NHW (NH * NW)

__device__ __forceinline__ float leaky(float v) {
    return v >= 0.0f ? v : 0.01f * v;
}

// ---------------------------------------------------------------------------
// Stage 1: per-row GEMM (32 classes x 608 px, K=64) via v_wmma_f32_16x16x4_f32
// One wave (32 lanes) per (row h, 16-wide w tile). Two 16x16 accumulator tiles
// cover classes 0-15 and 16-31. Then bias + leaky + argmax -> global index.
// ---------------------------------------------------------------------------
__global__ __launch_bounds__(32)
void stage1_wmma_kernel(const float* __restrict__ x,
                        const float* __restrict__ w1,
                        const float* __restrict__ b1,
                        int* __restrict__ inds) {
    const int h     = blockIdx.y;
    const int wbase = blockIdx.x * 16;
    const int lane  = threadIdx.x;          // 0..31, full wave active
    const int n     = lane & 15;            // column within tile / M row for A
    const int khalf = lane >> 4;            // 0: K+0..1, 1: K+2..3

    const float* __restrict__ w1h = w1 + (size_t)h * NS1 * NC;  // (32 x 64) row-major
    const float* __restrict__ xh  = x + (size_t)h * NW + wbase; // x[c][h][w]

    v8f acc_lo = {};   // classes 0..15
    v8f acc_hi = {};   // classes 16..31

#pragma unroll
    for (int k0 = 0; k0 < NC; k0 += 4) {
        const int ka = k0 + khalf * 2;
        // A (16x4 fp32): lane m=n holds (K=ka, ka+1)
        v2f a_lo, a_hi, b;
        a_lo.x = w1h[n * NC + ka];
        a_lo.y = w1h[n * NC + ka + 1];
        a_hi.x = w1h[(n + 16) * NC + ka];
        a_hi.y = w1h[(n + 16) * NC + ka + 1];
        // B (4x16 fp32): lane n holds x[c=ka..ka+1][h][wbase+n]
        b.x = xh[(size_t)ka * NHW + n];
        b.y = xh[(size_t)(ka + 1) * NHW + n];

        acc_lo = __builtin_amdgcn_wmma_f32_16x16x4_f32(
            false, a_lo, false, b, (short)0, acc_lo, false, false);
        acc_hi = __builtin_amdgcn_wmma_f32_16x16x4_f32(
            false, a_hi, false, b, (short)0, acc_hi, false, false);
    }

    // D layout: VGPR v -> class v (lanes 0-15) / v+8 (lanes 16-31), lane&15 -> w.
    const float* __restrict__ b1h = b1 + (size_t)h * NS1;
    const int cb = khalf * 8;
    float best  = -3.4e38f;
    int   bestk = 0;
#pragma unroll
    for (int v = 0; v < 8; ++v) {
        float vl = leaky(acc_lo[v] + b1h[cb + v]);
        if (vl > best) { best = vl; bestk = cb + v; }
        float vh = leaky(acc_hi[v] + b1h[16 + cb + v]);
        if (vh > best) { best = vh; bestk = 16 + cb + v; }
    }
    // merge lane pair (n, n+16): each holds 16 of the 32 classes for pixel n
    float ob = __shfl_xor(best, 16, 32);
    int   ok = __shfl_xor(bestk, 16, 32);
    if (ob > best || (ob == best && ok < bestk)) { best = ob; bestk = ok; }

    if (lane < 16) {
        inds[(size_t)h * NW + wbase + n] = h * NS1 + bestk;  // global row-offset index
    }
}

// ---------------------------------------------------------------------------
// Stages 2+3 fused: per-pixel gathered matvecs (L2-resident w2/w3).
// Block = 256 threads (8 waves) handles a 32-pixel tile of one row.
// x tile (64 x 32) staged in LDS with +1 padding (conflict-free column reads).
// One wave per pixel: lanes (k=lane&15, chunk=lane>>4) cover the 64x16 matvec
// with fully coalesced w2 row reads; argmax via xor-shuffle; then 64x2 matvec.
// ---------------------------------------------------------------------------
#define WTILE 32

__global__ __launch_bounds__(256)
void stage23_kernel(const float* __restrict__ x,
                    const float* __restrict__ w2,
                    const float* __restrict__ b2,
                    const float* __restrict__ w3,
                    const float* __restrict__ b3,
                    const int* __restrict__ inds,
                    float* __restrict__ x_out,
                    float* __restrict__ mask) {
    __shared__ float xs[NC][WTILE + 1];

    const int h     = blockIdx.y;
    const int wbase = blockIdx.x * WTILE;
    const int tid   = threadIdx.x;

    // Cooperative coalesced load of x[c][h][wbase..wbase+31] tile
    for (int i = tid; i < NC * WTILE; i += 256) {
        const int c = i / WTILE;
        const int w = i % WTILE;
        xs[c][w] = x[(size_t)c * NHW + (size_t)h * NW + wbase + w];
    }
    __syncthreads();

    const int wave  = tid >> 5;
    const int lane  = tid & 31;
    const int k     = lane & 15;   // output class within S2
    const int chunk = lane >> 4;   // which 32-channel half

#pragma unroll
    for (int pp = 0; pp < WTILE / 8; ++pp) {
        const int wloc = pp * 8 + wave;
        const int pix  = h * NW + wbase + wloc;
        const int ind  = inds[pix];                       // uniform per wave -> SGPR

        // stage 2: (1x64) . (64x16) matvec, w2 rows read coalesced
        const float* __restrict__ w2p = w2 + (size_t)ind * (NC * NS2);
        float acc = 0.0f;
#pragma unroll
        for (int cl = 0; cl < 32; ++cl) {
            const int c = chunk * 32 + cl;
            acc += xs[c][wloc] * w2p[c * NS2 + k];
        }
        acc += __shfl_xor(acc, 16, 32);                  // combine channel halves
        float cls = leaky(acc + b2[(size_t)ind * NS2 + k]);

        // argmax over 16 classes (first-occurrence tie-break, like jnp.argmax)
        int bk = k;
#pragma unroll
        for (int off = 8; off >= 1; off >>= 1) {
            float ov = __shfl_xor(cls, off, 32);
            int   oi = __shfl_xor(bk, off, 32);
            if (ov > cls || (ov == cls && oi < bk)) { cls = ov; bk = oi; }
        }
        const long long ind2 = (long long)ind * NS2 + bk;

        // stage 3: (1x64) . (64x2) matvec, lane covers channels 2L, 2L+1
        const float* __restrict__ w3p = w3 + (size_t)ind2 * (NC * 2);
        const float x0 = xs[2 * lane][wloc];
        const float x1 = xs[2 * lane + 1][wloc];
        float r0 = x0 * w3p[4 * lane]     + x1 * w3p[4 * lane + 2];
        float r1 = x0 * w3p[4 * lane + 1] + x1 * w3p[4 * lane + 3];
#pragma unroll
        for (int off = 16; off >= 1; off >>= 1) {
            r0 += __shfl_xor(r0, off, 32);
            r1 += __shfl_xor(r1, off, 32);
        }

        if (lane == 0) {
            r0 += b3[(size_t)ind2 * 2 + 0];
            r1 += b3[(size_t)ind2 * 2 + 1];
            const int loc = (int)(ind2 - (long long)h * NS1 * NS2); // [0, S1*S2)
            x_out[pix] = ((float)loc + r0) * 0.5f;   // scale = (1/S1)*S2
            mask[pix]  = leaky(r1);
        }
    }
}

extern "C" void kernel_launch(void* const* d_in, const int* in_sizes, int n_in,
                              void* d_out, int out_size, void* d_ws, size_t ws_size,
                              hipStream_t stream) {
    const float* x  = (const float*)d_in[0];
    const float* w1 = (const float*)d_in[1];
    const float* b1 = (const float*)d_in[2];
    const float* w2 = (const float*)d_in[3];
    const float* b2 = (const float*)d_in[4];
    const float* w3 = (const float*)d_in[5];
    const float* b3 = (const float*)d_in[6];

    float* x_out = (float*)d_out;            // (H*W) then mask (H*W)
    float* mask  = x_out + NHW;
    int*   inds  = (int*)d_ws;               // N int32 = 1.09 MB scratch

    dim3 g1(NW / 16, NH);                    // 38 x 448 waves
    stage1_wmma_kernel<<<g1, 32, 0, stream>>>(x, w1, b1, inds);

    dim3 g2(NW / WTILE, NH);                 // 19 x 448 blocks of 256
    stage23_kernel<<<g2, 256, 0, stream>>>(x, w2, b2, w3, b3, inds, x_out, mask);
}